// BidirectionalMamba_17832704213070
// MI455X (gfx1250) — compile-verified
//
#include <hip/hip_runtime.h>
#include <math.h>

// ---- model dims (fixed by reference) ----
#define NLY   2
#define DMOD  256
#define DIN   512
#define FFNH  1024
#define VOC   1024
#define LSEQ  1024
#define RRK   16
#define NST   16
#define KCV   4
#define BSZ   2
#define ROWS  (BSZ * LSEQ)   // 2048
#define EPSF  1e-5f

typedef __attribute__((ext_vector_type(2))) float v2f;
typedef __attribute__((ext_vector_type(8))) float v8f;

__device__ __forceinline__ float silu_f(float x) { return x / (1.0f + __expf(-x)); }
__device__ __forceinline__ float softplus_f(float x) {
  return (x > 20.0f) ? x : log1pf(__expf(x));
}
__device__ __forceinline__ int rev_row(int r) {
  return (r & ~(LSEQ - 1)) | ((LSEQ - 1) - (r & (LSEQ - 1)));
}

// =====================================================================
// fp32 WMMA GEMM: C[M,N] = A[M,K] * B[K,N]  (+ epilogue)
// One 16 x (16*NT) strip per wave32; A fragment reused across NT tiles.
// All strides / trip counts compile-time -> immediate-offset loads.
// EPI: 0 none, 1 +bias, 2 +bias,relu, 3 +bias,softplus
// REVA/REVC: reverse sequence index on A-read / C-write (per batch)
// =====================================================================
template <int KDIM, int LDA, int LDB, int NT, int EPI, bool REVA, bool REVC>
__global__ __launch_bounds__(256) void wmma_gemm_f32(
    const float* __restrict__ A, const float* __restrict__ Bmat,
    float* __restrict__ C, const float* __restrict__ bias,
    int M, int Ncol)
{
  const int lane    = threadIdx.x & 31;
  const int wave    = threadIdx.x >> 5;
  const int stripsN = Ncol / (16 * NT);
  const int strip   = blockIdx.x * 8 + wave;
  if (strip >= (M >> 4) * stripsN) return;        // wave-uniform guard
  const int tm   = strip / stripsN;
  const int col0 = (strip - tm * stripsN) * (16 * NT);

  const int half = lane >> 4;      // 0: lanes 0-15, 1: lanes 16-31
  const int l15  = lane & 15;

  // A fragment (16x4 f32): lanes 0-15 rows hold K pair {0,1}, lanes 16-31 {2,3}
  int arow = tm * 16 + l15;
  if (REVA) arow = rev_row(arow);
  const float* Aptr = A + (size_t)arow * LDA + half * 2;
  // B fragment (4x16 f32): K rows striped across the VGPR pair
  const float* Bptr = Bmat + (size_t)(half * 2) * LDB + col0 + l15;

  v8f acc[NT];
#pragma unroll
  for (int t = 0; t < NT; ++t) acc[t] = (v8f){0.f,0.f,0.f,0.f,0.f,0.f,0.f,0.f};

#pragma unroll 4
  for (int k0 = 0; k0 < KDIM; k0 += 4) {
    v2f a = *(const v2f*)(Aptr + k0);             // 8B-aligned, consecutive K
    if (KDIM > 64) __builtin_prefetch(Bptr + (size_t)(k0 + 64) * LDB, 0, 0);
#pragma unroll
    for (int t = 0; t < NT; ++t) {
      v2f b;
      b.x = Bptr[(size_t)k0 * LDB + t * 16];
      b.y = Bptr[(size_t)(k0 + 1) * LDB + t * 16];
      acc[t] = __builtin_amdgcn_wmma_f32_16x16x4_f32(
          /*neg_a=*/false, a, /*neg_b=*/false, b,
          /*c_mod=*/(short)0, acc[t], /*reuse_a=*/false, /*reuse_b=*/false);
    }
  }

  // C/D layout: VGPR r -> M = r (lanes 0-15) or r+8 (lanes 16-31); N = lane&15
#pragma unroll
  for (int t = 0; t < NT; ++t) {
    const int col = col0 + t * 16 + l15;
    const float bv = (EPI >= 1) ? bias[col] : 0.0f;
#pragma unroll
    for (int r = 0; r < 8; ++r) {
      int row = tm * 16 + half * 8 + r;
      float v = acc[t][r];
      if (EPI >= 1) v += bv;
      if (EPI == 2) v = fmaxf(v, 0.0f);
      if (EPI == 3) v = softplus_f(v);
      int orow = REVC ? rev_row(row) : row;
      C[(size_t)orow * Ncol + col] = v;
    }
  }
}

// ---------------------------------------------------------------------
// x = emb[ids] * sqrt(D) + pos
// ---------------------------------------------------------------------
__global__ __launch_bounds__(256) void embed_kernel(
    const int* __restrict__ ids, const float* __restrict__ emb,
    const float* __restrict__ pos, float* __restrict__ x)
{
  int i = blockIdx.x * 256 + threadIdx.x;
  if (i >= ROWS * DMOD) return;
  int d = i & (DMOD - 1);
  int row = i / DMOD;
  int l = row & (LSEQ - 1);
  int tok = ids[row];
  x[i] = emb[(size_t)tok * DMOD + d] * 16.0f + pos[(size_t)l * DMOD + d];
}

// ---------------------------------------------------------------------
// causal depthwise conv (K=4) over xi = xz[:, :DIN], + bias, + SiLU
// ---------------------------------------------------------------------
__global__ __launch_bounds__(256) void conv_silu_kernel(
    const float* __restrict__ xz, const float* __restrict__ cw,
    const float* __restrict__ cb, float* __restrict__ xc)
{
  int i = blockIdx.x * 256 + threadIdx.x;
  if (i >= ROWS * DIN) return;
  int c = i & (DIN - 1);
  int row = i / DIN;
  int b = row / LSEQ;
  int l = row - b * LSEQ;
  float acc = cb[c];
#pragma unroll
  for (int k = 0; k < KCV; ++k) {
    int ls = l + k - (KCV - 1);
    if (ls >= 0)
      acc += cw[c * KCV + k] * xz[((size_t)(b * LSEQ + ls)) * (2 * DIN) + c];
  }
  xc[i] = silu_f(acc);
}

// ---------------------------------------------------------------------
// selective scan: 4 lanes per (b,d), 4 states per lane (quad q = lane&3).
// h = h*exp(dt*A) + dt*B_t*xc ; y_t = <h, C_t> (finished via shfl_xor) ;
// out = (y + Dsk*xc) * silu(z)
// ---------------------------------------------------------------------
__global__ __launch_bounds__(256) void scan_kernel(
    const float* __restrict__ delta, const float* __restrict__ xc,
    const float* __restrict__ xdb, const float* __restrict__ xz,
    const float* __restrict__ A_log, const float* __restrict__ Dskip,
    float* __restrict__ y2)
{
  int gid = blockIdx.x * 256 + threadIdx.x;   // [0, 4*DIN)
  int b = blockIdx.y;
  int q = gid & 3;           // state quad: states q*4 .. q*4+3
  int d = gid >> 2;          // channel
  if (d >= DIN) return;

  float Ac[4], h[4];
#pragma unroll
  for (int j = 0; j < 4; ++j) {
    Ac[j] = -__expf(A_log[d * NST + q * 4 + j]);
    h[j] = 0.0f;
  }
  const float dsk = Dskip[d];

  for (int t = 0; t < LSEQ; ++t) {
    size_t row = (size_t)b * LSEQ + t;
    float dv  = delta[row * DIN + d];
    float xcv = xc[row * DIN + d];
    const float4 Bv = *(const float4*)(xdb + row * 48 + 16 + q * 4);
    const float4 Cv = *(const float4*)(xdb + row * 48 + 32 + q * 4);
    float y;
    {
      float dA0 = __expf(dv * Ac[0]), dA1 = __expf(dv * Ac[1]);
      float dA2 = __expf(dv * Ac[2]), dA3 = __expf(dv * Ac[3]);
      float dbx = dv * xcv;
      h[0] = h[0] * dA0 + dbx * Bv.x;
      h[1] = h[1] * dA1 + dbx * Bv.y;
      h[2] = h[2] * dA2 + dbx * Bv.z;
      h[3] = h[3] * dA3 + dbx * Bv.w;
      y = h[0] * Cv.x + h[1] * Cv.y + h[2] * Cv.z + h[3] * Cv.w;
    }
    y += __shfl_xor(y, 1);
    y += __shfl_xor(y, 2);
    if (q == 0) {
      float zv = xz[row * (2 * DIN) + DIN + d];
      y2[row * DIN + d] = (y + dsk * xcv) * silu_f(zv);
    }
  }
}

// ---------------------------------------------------------------------
// batchnorm stats over (B,L) per channel -> scale/shift
// ---------------------------------------------------------------------
__global__ __launch_bounds__(256) void bn_stats_kernel(
    const float* __restrict__ src, const float* __restrict__ g,
    const float* __restrict__ bb, float* __restrict__ scale,
    float* __restrict__ shift)
{
  __shared__ float s1[256], s2[256];
  int c = blockIdx.x;
  float s = 0.f, q = 0.f;
  for (int r = threadIdx.x; r < ROWS; r += 256) {
    float v = src[(size_t)r * DMOD + c];
    s += v; q += v * v;
  }
  s1[threadIdx.x] = s; s2[threadIdx.x] = q;
  __syncthreads();
  for (int o = 128; o > 0; o >>= 1) {
    if (threadIdx.x < o) { s1[threadIdx.x] += s1[threadIdx.x + o]; s2[threadIdx.x] += s2[threadIdx.x + o]; }
    __syncthreads();
  }
  if (threadIdx.x == 0) {
    float m = s1[0] * (1.0f / ROWS);
    float var = s2[0] * (1.0f / ROWS) - m * m;
    float sc = g[c] * rsqrtf(var + EPSF);
    scale[c] = sc;
    shift[c] = bb[c] - m * sc;
  }
}

// ---------------------------------------------------------------------
// fused: tf = x + BN(mf); tb = x + BN(mb); out = LN(tf) + LN(tb)
// one block (256 threads) per row; D == 256 == blockDim
// ---------------------------------------------------------------------
__global__ __launch_bounds__(256) void fuse_ln_kernel(
    const float* __restrict__ x, const float* __restrict__ mf,
    const float* __restrict__ mb,
    const float* __restrict__ scF, const float* __restrict__ shF,
    const float* __restrict__ scB, const float* __restrict__ shB,
    const float* __restrict__ lgF, const float* __restrict__ lbF,
    const float* __restrict__ lgB, const float* __restrict__ lbB,
    float* __restrict__ out)
{
  __shared__ float r1[256], r2[256];
  int row = blockIdx.x, c = threadIdx.x;
  size_t idx = (size_t)row * DMOD + c;
  float xv = x[idx];
  float tf = xv + mf[idx] * scF[c] + shF[c];
  float tb = xv + mb[idx] * scB[c] + shB[c];

  r1[c] = tf; r2[c] = tf * tf;
  __syncthreads();
  for (int o = 128; o > 0; o >>= 1) {
    if (c < o) { r1[c] += r1[c + o]; r2[c] += r2[c + o]; }
    __syncthreads();
  }
  float mF = r1[0] * (1.0f / DMOD);
  float vF = r2[0] * (1.0f / DMOD) - mF * mF;
  __syncthreads();
  float fo = (tf - mF) * rsqrtf(vF + EPSF) * lgF[c] + lbF[c];

  r1[c] = tb; r2[c] = tb * tb;
  __syncthreads();
  for (int o = 128; o > 0; o >>= 1) {
    if (c < o) { r1[c] += r1[c + o]; r2[c] += r2[c + o]; }
    __syncthreads();
  }
  float mB = r1[0] * (1.0f / DMOD);
  float vB = r2[0] * (1.0f / DMOD) - mB * mB;
  __syncthreads();
  float bo = (tb - mB) * rsqrtf(vB + EPSF) * lgB[c] + lbB[c];

  out[idx] = fo + bo;
}

// ---------------------------------------------------------------------
// x += BN(f2)
// ---------------------------------------------------------------------
__global__ __launch_bounds__(256) void bn_residual_kernel(
    const float* __restrict__ f2, const float* __restrict__ sc,
    const float* __restrict__ sh, float* __restrict__ x)
{
  int i = blockIdx.x * 256 + threadIdx.x;
  if (i >= ROWS * DMOD) return;
  int c = i & (DMOD - 1);
  x[i] += f2[i] * sc[c] + sh[c];
}

// =====================================================================
static inline int gemm_blocks(int M, int N, int NT) {
  int strips = (M / 16) * (N / (16 * NT));
  return (strips + 7) / 8;
}

extern "C" void kernel_launch(void* const* d_in, const int* in_sizes, int n_in,
                              void* d_out, int out_size, void* d_ws, size_t ws_size,
                              hipStream_t stream) {
  (void)in_sizes; (void)n_in; (void)out_size; (void)ws_size;
  const int*   ids     = (const int*)  d_in[0];
  const float* emb     = (const float*)d_in[1];
  const float* pos     = (const float*)d_in[2];
  const float* Win     = (const float*)d_in[3];
  const float* conv_w  = (const float*)d_in[4];
  const float* conv_b  = (const float*)d_in[5];
  const float* Wx      = (const float*)d_in[6];
  const float* Wdt     = (const float*)d_in[7];
  const float* bdt     = (const float*)d_in[8];
  const float* A_log   = (const float*)d_in[9];
  const float* Dskip   = (const float*)d_in[10];
  const float* Wout    = (const float*)d_in[11];
  const float* bout    = (const float*)d_in[12];
  const float* bn_g    = (const float*)d_in[13];
  const float* bn_b    = (const float*)d_in[14];
  const float* ln_g    = (const float*)d_in[15];
  const float* ln_b    = (const float*)d_in[16];
  const float* ff_w1   = (const float*)d_in[17];
  const float* ff_b1   = (const float*)d_in[18];
  const float* ff_w2   = (const float*)d_in[19];
  const float* ff_b2   = (const float*)d_in[20];
  const float* final_w = (const float*)d_in[21];
  const float* final_b = (const float*)d_in[22];

  // workspace carve-up (~32 MB of fp32)
  float* p = (float*)d_ws;
  float* x_buf = p;  p += (size_t)ROWS * DMOD;
  float* xz    = p;  p += (size_t)ROWS * 2 * DIN;   // reused as FFN hidden
  float* xc    = p;  p += (size_t)ROWS * DIN;
  float* xdb   = p;  p += (size_t)ROWS * 48;
  float* delt  = p;  p += (size_t)ROWS * DIN;
  float* y2    = p;  p += (size_t)ROWS * DIN;
  float* mf    = p;  p += (size_t)ROWS * DMOD;
  float* mb    = p;  p += (size_t)ROWS * DMOD;
  float* ffin  = p;  p += (size_t)ROWS * DMOD;
  float* f2    = p;  p += (size_t)ROWS * DMOD;
  float* scF   = p;  p += DMOD;
  float* shF   = p;  p += DMOD;
  float* scB   = p;  p += DMOD;
  float* shB   = p;  p += DMOD;
  float* h1    = xz;   // FFN hidden aliases xz (2048 x 1024)

  // embedding + positional
  embed_kernel<<<(ROWS * DMOD + 255) / 256, 256, 0, stream>>>(ids, emb, pos, x_buf);

  for (int i = 0; i < NLY; ++i) {
    for (int dir = 0; dir < 2; ++dir) {
      int pd = i * 2 + dir;
      const float* Win_p  = Win    + (size_t)pd * DMOD * 2 * DIN;
      const float* cw_p   = conv_w + (size_t)pd * DIN * KCV;
      const float* cb_p   = conv_b + (size_t)pd * DIN;
      const float* Wx_p   = Wx     + (size_t)pd * DIN * 48;
      const float* Wdt_p  = Wdt    + (size_t)pd * RRK * DIN;
      const float* bdt_p  = bdt    + (size_t)pd * DIN;
      const float* Al_p   = A_log  + (size_t)pd * DIN * NST;
      const float* Dsk_p  = Dskip  + (size_t)pd * DIN;
      const float* Wout_p = Wout   + (size_t)pd * DIN * DMOD;
      const float* bout_p = bout   + (size_t)pd * DMOD;
      float* mout = dir ? mb : mf;

      // xz = x(rev?) @ Win              [2048,256] x [256,1024]
      if (dir)
        wmma_gemm_f32<DMOD, DMOD, 2*DIN, 4, 0, true,  false>
          <<<gemm_blocks(ROWS, 2*DIN, 4), 256, 0, stream>>>(x_buf, Win_p, xz, nullptr, ROWS, 2*DIN);
      else
        wmma_gemm_f32<DMOD, DMOD, 2*DIN, 4, 0, false, false>
          <<<gemm_blocks(ROWS, 2*DIN, 4), 256, 0, stream>>>(x_buf, Win_p, xz, nullptr, ROWS, 2*DIN);

      // xc = silu(conv(xi) + cb)
      conv_silu_kernel<<<(ROWS * DIN + 255) / 256, 256, 0, stream>>>(xz, cw_p, cb_p, xc);

      // xdb = xc @ Wx                   [2048,512] x [512,48]  (3 tiles/wave)
      wmma_gemm_f32<DIN, DIN, 48, 3, 0, false, false>
        <<<gemm_blocks(ROWS, 48, 3), 256, 0, stream>>>(xc, Wx_p, xdb, nullptr, ROWS, 48);

      // delta = softplus(xdb[:, :16] @ Wdt + bdt)   [2048,16] x [16,512]
      wmma_gemm_f32<RRK, 48, DIN, 4, 3, false, false>
        <<<gemm_blocks(ROWS, DIN, 4), 256, 0, stream>>>(xdb, Wdt_p, delt, bdt_p, ROWS, DIN);

      // selective scan + skip + silu(z) gate
      scan_kernel<<<dim3((4 * DIN) / 256, BSZ), 256, 0, stream>>>(
          delt, xc, xdb, xz, Al_p, Dsk_p, y2);

      // mout = y2 @ Wout + bout (store rev?)  [2048,512] x [512,256]
      if (dir)
        wmma_gemm_f32<DIN, DIN, DMOD, 4, 1, false, true>
          <<<gemm_blocks(ROWS, DMOD, 4), 256, 0, stream>>>(y2, Wout_p, mout, bout_p, ROWS, DMOD);
      else
        wmma_gemm_f32<DIN, DIN, DMOD, 4, 1, false, false>
          <<<gemm_blocks(ROWS, DMOD, 4), 256, 0, stream>>>(y2, Wout_p, mout, bout_p, ROWS, DMOD);
    }

    // batchnorm stats for fwd/bwd mamba outputs (BN is reversal-invariant)
    bn_stats_kernel<<<DMOD, 256, 0, stream>>>(mf, bn_g + (size_t)(i * 3 + 0) * DMOD,
                                              bn_b + (size_t)(i * 3 + 0) * DMOD, scF, shF);
    bn_stats_kernel<<<DMOD, 256, 0, stream>>>(mb, bn_g + (size_t)(i * 3 + 1) * DMOD,
                                              bn_b + (size_t)(i * 3 + 1) * DMOD, scB, shB);
    // ffin = LN(x + BN(mf)) + LN(x + BN(mb))
    fuse_ln_kernel<<<ROWS, 256, 0, stream>>>(x_buf, mf, mb, scF, shF, scB, shB,
                                             ln_g + (size_t)(i * 2 + 0) * DMOD,
                                             ln_b + (size_t)(i * 2 + 0) * DMOD,
                                             ln_g + (size_t)(i * 2 + 1) * DMOD,
                                             ln_b + (size_t)(i * 2 + 1) * DMOD, ffin);
    // FFN
    wmma_gemm_f32<DMOD, DMOD, FFNH, 4, 2, false, false>
      <<<gemm_blocks(ROWS, FFNH, 4), 256, 0, stream>>>(
        ffin, ff_w1 + (size_t)i * DMOD * FFNH, h1, ff_b1 + (size_t)i * FFNH, ROWS, FFNH);
    wmma_gemm_f32<FFNH, FFNH, DMOD, 4, 1, false, false>
      <<<gemm_blocks(ROWS, DMOD, 4), 256, 0, stream>>>(
        h1, ff_w2 + (size_t)i * FFNH * DMOD, f2, ff_b2 + (size_t)i * DMOD, ROWS, DMOD);
    // x += BN(f2)
    bn_stats_kernel<<<DMOD, 256, 0, stream>>>(f2, bn_g + (size_t)(i * 3 + 2) * DMOD,
                                              bn_b + (size_t)(i * 3 + 2) * DMOD, scF, shF);
    bn_residual_kernel<<<(ROWS * DMOD + 255) / 256, 256, 0, stream>>>(f2, scF, shF, x_buf);
  }

  // logits = x @ final_w + final_b   [2048,256] x [256,1024]
  wmma_gemm_f32<DMOD, DMOD, VOC, 4, 1, false, false>
    <<<gemm_blocks(ROWS, VOC, 4), 256, 0, stream>>>(
      x_buf, final_w, (float*)d_out, final_b, ROWS, VOC);
}